// KLBeta_37881611550799
// MI455X (gfx1250) — compile-verified
//
#include <hip/hip_runtime.h>

#define T_DIM 1024
#define NBLK  32      // number of b-blocks (4096 / 128)
#define TTILE 32      // timesteps per block (one lane per t)
#define NWARP 8       // 256 threads = 8 wave32
#define BITER 16      // b-rows per warp per block (128 / 8)

typedef __attribute__((ext_vector_type(2))) float v2f;
typedef __attribute__((ext_vector_type(8))) float v8f;
typedef int v2i __attribute__((vector_size(8)));   // int2 for the b64 builtin

#define AS1 __attribute__((address_space(1)))
#define AS3 __attribute__((address_space(3)))

// ---------------- CDNA5 async global->LDS helpers ----------------
#if __has_builtin(__builtin_amdgcn_global_load_async_to_lds_b32)
__device__ __forceinline__ void g2l_b32(const float* g, float* l) {
  __builtin_amdgcn_global_load_async_to_lds_b32((AS1 int*)(g), (AS3 int*)(l), 0, 0);
}
#else
__device__ __forceinline__ void g2l_b32(const float* g, float* l) {
  asm volatile("global_load_async_to_lds_b32 %0, %1, off"
               :: "v"((unsigned)(size_t)(void*)l), "v"(g) : "memory");
}
#endif

#if __has_builtin(__builtin_amdgcn_global_load_async_to_lds_b64)
__device__ __forceinline__ void g2l_b64(const float* g, float* l) {
  __builtin_amdgcn_global_load_async_to_lds_b64((AS1 v2i*)(g), (AS3 v2i*)(l), 0, 0);
}
#else
__device__ __forceinline__ void g2l_b64(const float* g, float* l) {
  asm volatile("global_load_async_to_lds_b64 %0, %1, off"
               :: "v"((unsigned)(size_t)(void*)l), "v"(g) : "memory");
}
#endif

#if __has_builtin(__builtin_amdgcn_s_wait_asynccnt)
#define WAIT_ASYNC(n) __builtin_amdgcn_s_wait_asynccnt(n)
#else
#define WAIT_ASYNC(n) asm volatile("s_wait_asynccnt %0" :: "i"(n) : "memory")
#endif

// ---------------- math helpers: Stirling / digamma tails ----------------
// lnGamma(y) for y >= ~6:  (y-1/2)ln y - y + ln(2*pi)/2 + 1/(12y) - 1/(360y^3)
__device__ __forceinline__ float stirl(float y, float lny, float iy) {
  float i2 = iy * iy;
  return (y - 0.5f) * lny - y + 0.91893853320467274f
       + iy * (8.3333333333333329e-2f - i2 * 2.7777777777777779e-3f);
}
// digamma(y) for y >= ~6: ln y - 1/(2y) - 1/(12y^2) + 1/(120y^4) - 1/(252y^6)
__device__ __forceinline__ float psit(float lny, float iy) {
  float i2 = iy * iy;
  return lny - 0.5f * iy
       - i2 * (8.3333333333333329e-2f
       - i2 * (8.3333333333333332e-3f
       - i2 *  3.9682539682539680e-3f));
}
__device__ __forceinline__ float rcpf(float x) { return __builtin_amdgcn_rcpf(x); }

// ============ K1: per-(b-block, t) class counts ============
__global__ __launch_bounds__(256) void k_counts(const float* __restrict__ tgt,
                                                int* __restrict__ cntp) {
  const int tid = threadIdx.x, lane = tid & 31, w = tid >> 5;
  const int t  = blockIdx.x * TTILE + lane;
  const int b0 = blockIdx.y * (NWARP * BITER);
  int c0 = 0, c1 = 0, c2 = 0;
#pragma unroll 4
  for (int i = 0; i < BITER; ++i) {
    int b = b0 + w + NWARP * i;
    float z = tgt[((size_t)b * T_DIM + t) * 3 + 2];
    if (z > -__builtin_inff()) {      // padded rows are all -inf
      int d = (int)z;
      c0 += (d == 0); c1 += (d == 1); c2 += (d == 2);
    }
  }
  __shared__ int red[NWARP][TTILE][3];
  red[w][lane][0] = c0; red[w][lane][1] = c1; red[w][lane][2] = c2;
  __syncthreads();
  if (tid < TTILE) {
    int s0 = 0, s1 = 0, s2 = 0;
    for (int ww = 0; ww < NWARP; ++ww) {
      s0 += red[ww][tid][0]; s1 += red[ww][tid][1]; s2 += red[ww][tid][2];
    }
    size_t o = ((size_t)blockIdx.y * T_DIM + blockIdx.x * TTILE + tid) * 3;
    cntp[o + 0] = s0; cntp[o + 1] = s1; cntp[o + 2] = s2;
  }
}

// ============ K2: reduce counts -> weights[t][3], n_t[t] ============
__global__ __launch_bounds__(256) void k_weights(const int* __restrict__ cntp,
                                                 float* __restrict__ wts,
                                                 float* __restrict__ nt) {
  int t = blockIdx.x * 256 + threadIdx.x;
  int s0 = 0, s1 = 0, s2 = 0;
#pragma unroll 8
  for (int k = 0; k < NBLK; ++k) {
    size_t o = ((size_t)k * T_DIM + t) * 3;
    s0 += cntp[o]; s1 += cntp[o + 1]; s2 += cntp[o + 2];
  }
  float n = (float)(s0 + s1 + s2);
  wts[t * 3 + 0] = n / (3.0f * (float)s0);   // inf if count==0 (never gathered)
  wts[t * 3 + 1] = n / (3.0f * (float)s1);
  wts[t * 3 + 2] = n / (3.0f * (float)s2);
  nt[t] = n;
}

// ============ K3: async-pipelined KL + weighted per-t partial sums ============
__global__ __launch_bounds__(256) void k_kl(const float* __restrict__ lgt,
                                            const float* __restrict__ tgt,
                                            const float* __restrict__ wts,
                                            float* __restrict__ sump) {
  const int tid = threadIdx.x, lane = tid & 31, w = tid >> 5;
  const int t0 = blockIdx.x * TTILE;
  const int b0 = blockIdx.y * (NWARP * BITER);

  __shared__ float lds_w[TTILE * 3];
  __shared__ float red[NWARP][TTILE];
  // per warp, double-buffered: 96 target floats (32*3) + 64 logit floats (32*2)
  __shared__ float buf[NWARP][2][160];

  if (tid < TTILE * 3) lds_w[tid] = wts[t0 * 3 + tid];
  __syncthreads();

  // prologue: issue async loads for iteration 0
  {
    int b = b0 + w;
    const float* gt = tgt + ((size_t)b * T_DIM + t0) * 3 + lane * 3;
    const float* gl = lgt + ((size_t)b * T_DIM + t0) * 2 + lane * 2;
    float* lb = &buf[w][0][0];
    g2l_b32(gt + 0, lb + lane * 3 + 0);
    g2l_b32(gt + 1, lb + lane * 3 + 1);
    g2l_b32(gt + 2, lb + lane * 3 + 2);
    g2l_b64(gl, lb + 96 + lane * 2);
  }

  float acc = 0.0f;
  for (int i = 0; i < BITER; ++i) {
    const int cur = i & 1;
    if (i + 1 < BITER) {  // prefetch next slice into the other buffer
      int b = b0 + w + NWARP * (i + 1);
      const float* gt = tgt + ((size_t)b * T_DIM + t0) * 3 + lane * 3;
      const float* gl = lgt + ((size_t)b * T_DIM + t0) * 2 + lane * 2;
      float* lb = &buf[w][cur ^ 1][0];
      g2l_b32(gt + 0, lb + lane * 3 + 0);
      g2l_b32(gt + 1, lb + lane * 3 + 1);
      g2l_b32(gt + 2, lb + lane * 3 + 2);
      g2l_b64(gl, lb + 96 + lane * 2);
      WAIT_ASYNC(4);   // 4 outstanding = the prefetch; current buffer complete
    } else {
      WAIT_ASYNC(0);
    }

    const float* lb = &buf[w][cur][0];
    float tw = lb[lane * 3 + 0];
    float tc = lb[lane * 3 + 1];
    float tz = lb[lane * 3 + 2];
    if (tz > -__builtin_inff()) {
      float l0 = lb[96 + lane * 2 + 0];
      float l1 = lb[96 + lane * 2 + 1];
      // predictions
      float pw  = rcpf(1.0f + __expf(-l0));
      float pcv = fmaf(1023.0f, rcpf(1.0f + __expf(-l1)), 5.0f);
      // Beta params (all > 1)
      float ta  = fmaf(tc, tw, 1.0f);
      float tbv = tc - ta + 2.0f;
      float pa  = fmaf(pcv, pw, 1.0f);
      float pb  = pcv - pa + 2.0f;
      float sp  = tc  + 2.0f;          // ta+tbv
      float sq  = pcv + 2.0f;          // pa+pb
      // shift-by-5 so all Stirling/psi args >= 6
      float ya = ta + 5.0f, yb = tbv + 5.0f, yc = pa + 5.0f, yd = pb + 5.0f;
      float la = __logf(ya), lbg = __logf(yb), lc = __logf(yc), ld = __logf(yd);
      float lsp = __logf(sp), lsq = __logf(sq);
      float ia = rcpf(ya), ib = rcpf(yb), ic = rcpf(yc), id = rcpf(yd);
      float isp = rcpf(sp), isq = rcpf(sq);
      // lnGamma shift correction products (combined -> one log each side)
      float prod_t = (ta * tbv) * ((ta + 1.f) * (tbv + 1.f)) * ((ta + 2.f) * (tbv + 2.f))
                   * ((ta + 3.f) * (tbv + 3.f)) * ((ta + 4.f) * (tbv + 4.f));
      float prod_p = (pa * pb) * ((pa + 1.f) * (pb + 1.f)) * ((pa + 2.f) * (pb + 2.f))
                   * ((pa + 3.f) * (pb + 3.f)) * ((pa + 4.f) * (pb + 4.f));
      float lg_t = stirl(ya, la, ia) + stirl(yb, lbg, ib) - __logf(prod_t);
      float lg_p = stirl(yc, lc, ic) + stirl(yd, ld, id) - __logf(prod_p);
      // digamma via psi(x) = psi(x+5) - sum_{k=0..4} 1/(x+k)
      float ha = rcpf(ta) + rcpf(ta + 1.f) + rcpf(ta + 2.f) + rcpf(ta + 3.f) + rcpf(ta + 4.f);
      float hb = rcpf(tbv) + rcpf(tbv + 1.f) + rcpf(tbv + 2.f) + rcpf(tbv + 3.f) + rcpf(tbv + 4.f);
      float kl = lg_p - lg_t + stirl(sp, lsp, isp) - stirl(sq, lsq, isq)
               + (ta  - pa) * (psit(la,  ia) - ha)
               + (tbv - pb) * (psit(lbg, ib) - hb)
               + (sq  - sp) *  psit(lsp, isp);
      float wcl = lds_w[lane * 3 + (int)tz];
      acc = fmaf(wcl, kl, acc);
    }
  }

  red[w][lane] = acc;
  __syncthreads();
  if (tid < TTILE) {
    float s = 0.0f;
    for (int ww = 0; ww < NWARP; ++ww) s += red[ww][tid];
    sump[(size_t)blockIdx.y * T_DIM + t0 + tid] = s;
  }
}

// ============ K4: deterministic final reduce (WMMA ones-matrix sum) ============
__global__ __launch_bounds__(256) void k_final(const float* __restrict__ sump,
                                               const float* __restrict__ nt,
                                               float* __restrict__ out) {
  const int tid = threadIdx.x;
  __shared__ float vals[T_DIM];
#pragma unroll
  for (int tt = 0; tt < 4; ++tt) {
    int t = tid + 256 * tt;
    float s = 0.0f;
#pragma unroll 8
    for (int k = 0; k < NBLK; ++k) s += sump[(size_t)k * T_DIM + t];
    vals[t] = s / nt[t];   // loss_t
  }
  __syncthreads();
  if (tid >= 32) return;   // wave 0 only: EXEC all-ones for WMMA
#if __has_builtin(__builtin_amdgcn_wmma_f32_16x16x4_f32)
  v2f a = {1.0f, 1.0f};    // ones A (16x4) -> D[m][n] = column sums of B
  v8f c = {};
#pragma unroll
  for (int k = 0; k < T_DIM / 64; ++k) {
    v2f bb = *(const v2f*)&vals[k * 64 + tid * 2];   // 64 values per wave per step
    c = __builtin_amdgcn_wmma_f32_16x16x4_f32(false, a, false, bb,
                                              (short)0, c, false, false);
  }
  // total = sum over the 16 columns of row 0 (lanes 0..15, VGPR0)
  float v = c[0];
  for (int off = 8; off >= 1; off >>= 1) v += __shfl_xor(v, off, 32);
  if (tid == 0) out[0] = v * (1.0f / (float)T_DIM);
#else
  if (tid == 0) {
    float s = 0.0f;
    for (int t = 0; t < T_DIM; ++t) s += vals[t];
    out[0] = s * (1.0f / (float)T_DIM);
  }
#endif
}

// ============ launch ============
extern "C" void kernel_launch(void* const* d_in, const int* in_sizes, int n_in,
                              void* d_out, int out_size, void* d_ws, size_t ws_size,
                              hipStream_t stream) {
  (void)in_sizes; (void)n_in; (void)out_size; (void)ws_size;
  const float* logits  = (const float*)d_in[0];   // [B,T,2]
  const float* targets = (const float*)d_in[1];   // [B,T,3]
  float* out = (float*)d_out;

  char* ws = (char*)d_ws;
  int*   cntp = (int*)ws;                                     // [32][1024][3] int
  float* wts  = (float*)(ws + (size_t)NBLK * T_DIM * 3 * 4);  // [1024][3]
  float* nt   = (float*)((char*)wts + (size_t)T_DIM * 3 * 4); // [1024]
  float* sump = (float*)((char*)nt + (size_t)T_DIM * 4);      // [32][1024]

  dim3 grid(T_DIM / TTILE, NBLK);
  k_counts <<<grid, 256, 0, stream>>>(targets, cntp);
  k_weights<<<dim3(T_DIM / 256), 256, 0, stream>>>(cntp, wts, nt);
  k_kl     <<<grid, 256, 0, stream>>>(logits, targets, wts, sump);
  k_final  <<<1, 256, 0, stream>>>(sump, nt, out);
}